// Detr3DCrossAttention_68650757259988
// MI455X (gfx1250) — compile-verified
//
#include <hip/hip_runtime.h>
#include <math.h>

// Shapes (fixed by the reference)
#define Bn   2
#define Qn   900
#define BQ   1800      // B*Q
#define Cn   256
#define Nc   6
#define Ln   3

#define LDS_PITCH 260  // 256 + 4 pad: conflict-free 64-bank fragment reads

typedef __attribute__((ext_vector_type(2))) float v2f;
typedef __attribute__((ext_vector_type(8))) float v8f;

// ---------------------------------------------------------------------------
// Kernel 1: w[bq,18] = sigmoid((query+query_pos) @ W_attn + b_attn)
// ---------------------------------------------------------------------------
__global__ void attn_w_kernel(const float* __restrict__ query,
                              const float* __restrict__ qpos,
                              const float* __restrict__ W_attn,
                              const float* __restrict__ b_attn,
                              float* __restrict__ wbuf) {
  int bq = blockIdx.x;
  int j  = threadIdx.x;
  if (j >= 18) return;
  const float* qr = query + (size_t)bq * Cn;
  const float* pr = qpos  + (size_t)bq * Cn;
  float acc = b_attn[j];
  #pragma unroll 4
  for (int c = 0; c < Cn; ++c)
    acc = fmaf(qr[c] + pr[c], W_attn[c * 18 + j], acc);
  wbuf[bq * 18 + j] = 1.0f / (1.0f + __expf(-acc));
}

// ---------------------------------------------------------------------------
// Kernel 2: project reference points -> grid (gx,gy) + visibility mask
// ---------------------------------------------------------------------------
__global__ void project_kernel(const float* __restrict__ ref,
                               const float* __restrict__ lidar2img,
                               float* __restrict__ gridbuf,
                               float* __restrict__ maskbuf) {
  int i = blockIdx.x * blockDim.x + threadIdx.x;
  if (i >= Bn * Nc * Qn) return;
  int q = i % Qn;
  int n = (i / Qn) % Nc;
  int b = i / (Qn * Nc);
  const float* r = ref + ((size_t)b * Qn + q) * 3;
  float x = r[0] * 102.4f - 51.2f;          // PC_RANGE x
  float y = r[1] * 102.4f - 51.2f;          // PC_RANGE y
  float z = r[2] * 8.0f   - 5.0f;           // PC_RANGE z
  const float* M = lidar2img + ((size_t)b * Nc + n) * 16;
  float cx = fmaf(M[0], x, fmaf(M[1], y, fmaf(M[2],  z, M[3])));
  float cy = fmaf(M[4], x, fmaf(M[5], y, fmaf(M[6],  z, M[7])));
  float cz = fmaf(M[8], x, fmaf(M[9], y, fmaf(M[10], z, M[11])));
  float zc = fmaxf(cz, 1e-5f);
  float gx = (cx / zc * (1.0f / 1600.0f) - 0.5f) * 2.0f;
  float gy = (cy / zc * (1.0f / 928.0f)  - 0.5f) * 2.0f;
  gridbuf[2 * i]     = gx;
  gridbuf[2 * i + 1] = gy;
  bool m = (cz > 1e-5f) && (gx > -1.f) && (gx < 1.f) && (gy > -1.f) && (gy < 1.f);
  maskbuf[i] = m ? 1.0f : 0.0f;
}

// ---------------------------------------------------------------------------
// Kernel 3: PE hidden layer h[bq,c] = relu(isig(ref) @ W_pe1 + b_pe1)
// ---------------------------------------------------------------------------
__device__ __forceinline__ float isig(float x) {
  x = fminf(fmaxf(x, 0.f), 1.f);
  float x1 = fmaxf(x, 1e-5f);
  float x2 = fmaxf(1.f - x, 1e-5f);
  return __logf(x1 / x2);
}

__global__ void pos_hidden_kernel(const float* __restrict__ ref,
                                  const float* __restrict__ W_pe1,
                                  const float* __restrict__ b_pe1,
                                  float* __restrict__ hbuf) {
  int bq = blockIdx.x;
  int c  = threadIdx.x;
  const float* r = ref + (size_t)bq * 3;
  float r0 = isig(r[0]), r1 = isig(r[1]), r2 = isig(r[2]);
  float h = fmaf(r0, W_pe1[c], fmaf(r1, W_pe1[Cn + c], fmaf(r2, W_pe1[2 * Cn + c], b_pe1[c])));
  hbuf[(size_t)bq * Cn + c] = fmaxf(h, 0.f);
}

// ---------------------------------------------------------------------------
// Kernel 4: bilinear sample from 3 pyramid levels, weight, reduce over (n,l)
// One block per (b,q); 256 threads = channels. grid/mask are block-uniform.
// ---------------------------------------------------------------------------
__device__ __forceinline__ float bilinear(const float* __restrict__ f,
                                          int Hl, int Wl, float gx, float gy) {
  float x = (gx + 1.f) * 0.5f * (float)Wl - 0.5f;
  float y = (gy + 1.f) * 0.5f * (float)Hl - 0.5f;
  float x0f = floorf(x), y0f = floorf(y);
  int x0 = (int)x0f, y0 = (int)y0f;
  float wx1 = x - x0f, wx0 = 1.f - wx1;
  float wy1 = y - y0f, wy0 = 1.f - wy1;
  bool xv0 = (x0 >= 0) && (x0 < Wl);
  bool xv1 = (x0 + 1 >= 0) && (x0 + 1 < Wl);
  bool yv0 = (y0 >= 0) && (y0 < Hl);
  bool yv1 = (y0 + 1 >= 0) && (y0 + 1 < Hl);
  float s = 0.f;
  if (yv0) {
    const float* row = f + (size_t)y0 * Wl;
    if (xv0) s = fmaf(row[x0],     wx0 * wy0, s);
    if (xv1) s = fmaf(row[x0 + 1], wx1 * wy0, s);
  }
  if (yv1) {
    const float* row = f + (size_t)(y0 + 1) * Wl;
    if (xv0) s = fmaf(row[x0],     wx0 * wy1, s);
    if (xv1) s = fmaf(row[x0 + 1], wx1 * wy1, s);
  }
  return s;
}

__global__ void sample_fuse_kernel(const float* __restrict__ feat0,
                                   const float* __restrict__ feat1,
                                   const float* __restrict__ feat2,
                                   const float* __restrict__ gridbuf,
                                   const float* __restrict__ maskbuf,
                                   const float* __restrict__ wbuf,
                                   float* __restrict__ fused) {
  int bq = blockIdx.x;
  int c  = threadIdx.x;
  int b  = bq / Qn, q = bq % Qn;
  float acc = 0.f;
  #pragma unroll
  for (int n = 0; n < Nc; ++n) {
    int gi = (b * Nc + n) * Qn + q;
    float m = maskbuf[gi];              // uniform over the block
    if (m == 0.f) continue;
    float gx = gridbuf[2 * gi], gy = gridbuf[2 * gi + 1];
    const float* wrow = wbuf + bq * 18 + n * Ln;
    size_t bn = (size_t)(b * Nc + n);
    acc = fmaf(bilinear(feat0 + (bn * Cn + c) * 23200u, 116, 200, gx, gy), wrow[0], acc);
    acc = fmaf(bilinear(feat1 + (bn * Cn + c) * 5800u,   58, 100, gx, gy), wrow[1], acc);
    acc = fmaf(bilinear(feat2 + (bn * Cn + c) * 1450u,   29,  50, gx, gy), wrow[2], acc);
  }
  fused[(size_t)bq * Cn + c] = acc;
}

// ---------------------------------------------------------------------------
// Kernel 5: fused dual WMMA GEMM (full fp32 matrix path):
//   out = fused @ W_out  +  h @ W_pe2  +  (b_out + b_pe2)
// Block = 8 waves sharing one 16-row tile (tm); waves take 8 column tiles.
// A tiles (fused + h, 16x256 each) staged once in LDS (pitch 260 ->
// conflict-free b64 fragment reads across all 64 banks); B from global/L2.
// ---------------------------------------------------------------------------
__global__ void gemm_out_kernel(const float* __restrict__ Afused,
                                const float* __restrict__ Ah,
                                const float* __restrict__ Wout,
                                const float* __restrict__ Wpe2,
                                const float* __restrict__ bout,
                                const float* __restrict__ bpe2,
                                float* __restrict__ out) {
  __shared__ float lA1[16 * LDS_PITCH];
  __shared__ float lA2[16 * LDS_PITCH];

  int tm     = blockIdx.x >> 1;            // 0..112 row tile
  int tnBase = (blockIdx.x & 1) << 3;      // 0 or 8
  int t      = threadIdx.x;

  // Warm L2 for the shared weight matrices (global_prefetch_b8).
  __builtin_prefetch(Wout + (size_t)t * 32, 0, 1);
  __builtin_prefetch(Wpe2 + (size_t)t * 32, 0, 1);

  // Stage both A tiles: 256 threads x 16 floats (4x float4) each.
  {
    int r  = t >> 4;                       // 0..15 row within tile
    int cb = (t & 15) << 4;                // column base, 16-float chunk
    int grow = tm * 16 + r;
    if (grow > BQ - 1) grow = BQ - 1;      // clamp reads; tail rows never stored
    const float* g1 = Afused + (size_t)grow * Cn + cb;
    const float* g2 = Ah     + (size_t)grow * Cn + cb;
    float* s1 = lA1 + r * LDS_PITCH + cb;
    float* s2 = lA2 + r * LDS_PITCH + cb;
    #pragma unroll
    for (int i = 0; i < 4; ++i) {
      ((float4*)s1)[i] = ((const float4*)g1)[i];
      ((float4*)s2)[i] = ((const float4*)g2)[i];
    }
  }
  __syncthreads();

  int wave  = t >> 5;
  int lane  = t & 31;
  int tn    = tnBase + wave;
  int col   = tn * 16 + (lane & 15);
  int khalf = (lane >> 4) << 1;            // lanes 0-15 -> K+0/1 ; 16-31 -> K+2/3
  int arow  = (lane & 15) * LDS_PITCH;

  v8f acc = {};
  #pragma unroll 4
  for (int kk = 0; kk < Cn; kk += 4) {
    int kb = kk + khalf;
    v2f a1; a1.x = lA1[arow + kb]; a1.y = lA1[arow + kb + 1];
    v2f b1; b1.x = Wout[(size_t)kb * Cn + col]; b1.y = Wout[(size_t)(kb + 1) * Cn + col];
    acc = __builtin_amdgcn_wmma_f32_16x16x4_f32(false, a1, false, b1,
                                                (short)0, acc, false, false);
    v2f a2; a2.x = lA2[arow + kb]; a2.y = lA2[arow + kb + 1];
    v2f b2; b2.x = Wpe2[(size_t)kb * Cn + col]; b2.y = Wpe2[(size_t)(kb + 1) * Cn + col];
    acc = __builtin_amdgcn_wmma_f32_16x16x4_f32(false, a2, false, b2,
                                                (short)0, acc, false, false);
  }

  float bias = bout[col] + bpe2[col];
  int rbase = ((lane >> 4) << 3);          // D layout: row = v + 8*(lane>=16)
  if (tm < 112) {                          // full tile: uniform, unguarded stores
    float* orow = out + (size_t)(tm * 16 + rbase) * Cn + col;
    #pragma unroll
    for (int v = 0; v < 8; ++v)
      orow[(size_t)v * Cn] = acc[v] + bias;
  } else {                                 // partial tail tile (rows 1792..1807)
    #pragma unroll
    for (int v = 0; v < 8; ++v) {
      int r = tm * 16 + v + rbase;
      if (r < BQ) out[(size_t)r * Cn + col] = acc[v] + bias;
    }
  }
}

// ---------------------------------------------------------------------------
// Launch
// ---------------------------------------------------------------------------
extern "C" void kernel_launch(void* const* d_in, const int* in_sizes, int n_in,
                              void* d_out, int out_size, void* d_ws, size_t ws_size,
                              hipStream_t stream) {
  const float* query  = (const float*)d_in[0];
  const float* qpos   = (const float*)d_in[1];
  const float* feat0  = (const float*)d_in[2];
  const float* feat1  = (const float*)d_in[3];
  const float* feat2  = (const float*)d_in[4];
  const float* ref    = (const float*)d_in[5];
  const float* l2i    = (const float*)d_in[6];
  const float* W_attn = (const float*)d_in[7];
  const float* b_attn = (const float*)d_in[8];
  const float* W_out  = (const float*)d_in[9];
  const float* b_out  = (const float*)d_in[10];
  const float* W_pe1  = (const float*)d_in[11];
  const float* b_pe1  = (const float*)d_in[12];
  const float* W_pe2  = (const float*)d_in[13];
  const float* b_pe2  = (const float*)d_in[14];
  float* out = (float*)d_out;

  // Workspace layout (floats): w | grid | mask | fused | h  (~4 MB total)
  float* ws      = (float*)d_ws;
  float* wbuf    = ws;                     // 1800*18   = 32400
  float* gridbuf = ws + 32400;             // 10800*2   = 21600
  float* maskbuf = ws + 54000;             // 10800
  float* fusedb  = ws + 64800;             // 1800*256  = 460800
  float* hbuf    = ws + 525600;            // 1800*256  = 460800

  attn_w_kernel<<<BQ, 32, 0, stream>>>(query, qpos, W_attn, b_attn, wbuf);
  project_kernel<<<(Bn * Nc * Qn + 255) / 256, 256, 0, stream>>>(ref, l2i, gridbuf, maskbuf);
  pos_hidden_kernel<<<BQ, 256, 0, stream>>>(ref, W_pe1, b_pe1, hbuf);
  sample_fuse_kernel<<<BQ, 256, 0, stream>>>(feat0, feat1, feat2,
                                             gridbuf, maskbuf, wbuf, fusedb);
  // 113 row tiles x 2 half-column-ranges -> 226 blocks of 8 waves
  gemm_out_kernel<<<226, 256, 0, stream>>>(fusedb, hbuf, W_out, W_pe2, b_out, b_pe2, out);
}